// VAETransformer_FCGPFA_77369540870228
// MI455X (gfx1250) — compile-verified
//
#include <hip/hip_runtime.h>
#include <hip/hip_bf16.h>

// ---------------------------------------------------------------------------
// FCGPFA inner loop on MI455X (gfx1250).
//  - blocked sweep operator: hessian^{-1} in place, trailing rank-16 updates
//    run on V_WMMA_F32_16X16X4_F32 with LDS-staged panels (bulk of all FLOPs)
//  - mu @ inv_K GEMM also on WMMA
//  - FP32 throughout; working set ~65MB, L2-resident (192MB)
// ---------------------------------------------------------------------------

typedef float v2f __attribute__((ext_vector_type(2)));
typedef float v8f __attribute__((ext_vector_type(8)));

constexpr int TP    = 304;   // padded T (19 * 16)
constexpr int NB    = 19;    // block count (TP / 16)
constexpr int T_    = 300;
constexpr int NPAD_ = 50;
constexpr int M_    = 16;    // trials
constexpr int N_    = 96;    // neurons
constexpr int L_    = 8;     // latents
constexpr int S_    = 8;     // subspace
constexpr int NTAU  = 20;
constexpr int NTF   = 350;   // full time length of Y_raw
constexpr int MATS  = M_ * L_;            // 128 independent systems
constexpr size_t MSZ = (size_t)TP * TP;   // 92416 elements per matrix

__device__ __forceinline__ v8f wmma4(v2f a, v2f b, v8f c) {
  return __builtin_amdgcn_wmma_f32_16x16x4_f32(false, a, false, b, (short)0, c,
                                               false, false);
}

// ---- build padded (K + 1e-3 I) into matrix slot 0 -------------------------
__global__ void k_buildK(const float* __restrict__ Kin, float* __restrict__ A0) {
  int idx = blockIdx.x * blockDim.x + threadIdx.x;
  if (idx >= (int)MSZ) return;
  int i = idx / TP, j = idx % TP;
  float v;
  if (i < T_ && j < T_) v = Kin[i * T_ + j] + (i == j ? 0.001f : 0.0f);
  else                  v = (i == j) ? 1.0f : 0.0f;
  A0[idx] = v;
}

// ---- blocked in-place sweep of all 304 pivots: A -> -A^{-1} ---------------
// one workgroup (256 thr = 8 waves) per matrix. Per 16-wide block step:
//   S = -A_KK^{-1} (wave0, LDS)         P = (-S)*R          Q = C*(-S)
//   trailing A_IJ -= C_IK * P_KJ  via 4 chained WMMA per 16x16 tile
__global__ __launch_bounds__(256) void k_bsweep(float* __restrict__ Abase) {
  float* A = Abase + (size_t)blockIdx.x * MSZ;
  __shared__ float Cp[TP][17];     // column panel A[:, c0..c0+15] (padded)
  __shared__ float Rp[16][TP];     // row panel    A[c0..c0+15, :]
  __shared__ float Pp[16][TP];     // P = (-S) * R
  __shared__ float Sb[16][17];     // block pivot workspace
  __shared__ float rk[16], ck[16];
  const int tid  = threadIdx.x;
  const int lane = tid & 31, wave = tid >> 5;
  const int r = lane & 15, hi = lane >> 4;

  for (int kb = 0; kb < NB; ++kb) {
    const int c0 = kb * 16;
    // ---- stage row + column panels -------------------------------------
    for (int e = tid; e < 16 * TP; e += 256) {
      int rr = e / TP, jj = e % TP;
      Rp[rr][jj] = A[(size_t)(c0 + rr) * TP + jj];
      Cp[jj][rr] = A[(size_t)jj * TP + c0 + rr];
    }
    __syncthreads();

    // ---- wave0: sweep the 16x16 pivot block in LDS (S -> -A_KK^{-1}) ----
    if (wave == 0) {
      for (int e = lane; e < 256; e += 32) {
        int i = e >> 4, j = e & 15;
        Sb[i][j] = Rp[i][c0 + j];
      }
      __builtin_amdgcn_wave_barrier();
      for (int k = 0; k < 16; ++k) {
        if (lane < 16) { rk[lane] = Sb[k][lane]; ck[lane] = Sb[lane][k]; }
        __builtin_amdgcn_wave_barrier();
        const float dinv = 1.0f / rk[k];
        for (int e = lane; e < 256; e += 32) {
          int i = e >> 4, j = e & 15;
          if (i != k && j != k) Sb[i][j] = fmaf(-ck[i] * dinv, rk[j], Sb[i][j]);
        }
        __builtin_amdgcn_wave_barrier();
        if (lane < 16 && lane != k) {
          Sb[k][lane] = rk[lane] * dinv;
          Sb[lane][k] = ck[lane] * dinv;
        }
        if (lane == 0) Sb[k][k] = -dinv;
        __builtin_amdgcn_wave_barrier();
      }
    }
    __syncthreads();

    // ---- P = (-S) * R  (all 304 columns; block columns unused later) ----
    for (int e = tid; e < 16 * TP; e += 256) {
      int rr = e / TP, jj = e % TP;
      float acc = 0.0f;
#pragma unroll
      for (int p = 0; p < 16; ++p) acc = fmaf(-Sb[rr][p], Rp[p][jj], acc);
      Pp[rr][jj] = acc;
    }
    __syncthreads();

    // ---- trailing update: 18x18 tiles striped over 8 waves (WMMA) -------
    for (int tix = wave; tix < 18 * 18; tix += 8) {
      int ib = tix / 18, jb = tix % 18;
      if (ib >= kb) ++ib;
      if (jb >= kb) ++jb;
      const int i0 = ib * 16, j0 = jb * 16;
      v8f acc = {};
#pragma unroll
      for (int kk = 0; kk < 4; ++kk) {
        v2f a;  // A_IK fragment: lane = row, k = 4*kk + 2*hi + {0,1}
        a.x = Cp[i0 + r][4 * kk + 2 * hi + 0];
        a.y = Cp[i0 + r][4 * kk + 2 * hi + 1];
        v2f b;  // P fragment: lane = col
        b.x = Pp[4 * kk + 2 * hi + 0][j0 + r];
        b.y = Pp[4 * kk + 2 * hi + 1][j0 + r];
        acc = wmma4(a, b, acc);
      }
#pragma unroll
      for (int q = 0; q < 8; ++q) {
        const size_t o = (size_t)(i0 + q + 8 * hi) * TP + j0 + r;
        A[o] = A[o] - acc[q];
      }
    }

    // ---- write back panels (disjoint addresses from trailing tiles) ----
    for (int e = tid; e < 16 * TP; e += 256) {
      int rr = e / TP, jj = e % TP;
      if (jj < c0 || jj >= c0 + 16) {
        A[(size_t)(c0 + rr) * TP + jj] = Pp[rr][jj];          // A_KJ' = P
        float q = 0.0f;                                       // A_JK' = C*(-S)
#pragma unroll
        for (int p = 0; p < 16; ++p) q = fmaf(Cp[jj][p], -Sb[p][rr], q);
        A[(size_t)jj * TP + c0 + rr] = q;
      } else {
        A[(size_t)(c0 + rr) * TP + jj] = Sb[rr][jj - c0];     // A_KK' = S
      }
    }
    __syncthreads();
  }
}

// ---- invK = -sweep(K+1e-3 I), zero padding --------------------------------
__global__ void k_invk(const float* __restrict__ A0, float* __restrict__ invK) {
  int idx = blockIdx.x * blockDim.x + threadIdx.x;
  if (idx >= (int)MSZ) return;
  int i = idx / TP, j = idx % TP;
  invK[idx] = (i < T_ && j < T_) ? -A0[idx] : 0.0f;
}

// ---- fused FIR conv + threshold + subspace projection ---------------------
__global__ void k_weights(const float* __restrict__ Y, const float* __restrict__ kern,
                          const float* __restrict__ wproj, float* __restrict__ W) {
  int idx = blockIdx.x * blockDim.x + threadIdx.x;
  if (idx >= M_ * N_ * T_) return;
  int t = idx % T_, mn = idx / T_;
  const float* y = Y + (size_t)mn * NTF + NPAD_ + t;   // y[0] = Y[m,n,50+t]
  float G[S_];
#pragma unroll
  for (int s = 0; s < S_; ++s) G[s] = 0.0f;
  for (int tau = 1; tau <= NTAU; ++tau) {
    const float yv = y[-tau];
#pragma unroll
    for (int s = 0; s < S_; ++s) G[s] = fmaf(yv, kern[(tau - 1) * S_ + s], G[s]);
  }
#pragma unroll
  for (int s = 0; s < S_; ++s) if (fabsf(G[s]) < 1e-5f) G[s] = 0.0f;
#pragma unroll
  for (int l = 0; l < L_; ++l) {
    float w = 0.0f;
#pragma unroll
    for (int s = 0; s < S_; ++s) w = fmaf(G[s], wproj[s * L_ + l], w);
    W[((size_t)mn * L_ + l) * TP + t] = w;
  }
}

// ---- mu = 0 (incl. padding: WMMA A operand reads it), dh = 1.001 ----------
__global__ void k_init(float* __restrict__ mu, float* __restrict__ dh) {
  int idx = blockIdx.x * blockDim.x + threadIdx.x;
  if (idx >= MATS * TP) return;
  mu[idx] = 0.0f;
  dh[idx] = 1.001f;   // diag(K + 1e-3 I): exp(0) + 1e-3
}

// ---- lambda[m,n,t] = exp(sum_l W*mu + bias + 0.5*sum_l dh*W^2) ------------
__global__ void k_lambda(const float* __restrict__ W, const float* __restrict__ mu,
                         const float* __restrict__ dh, const float* __restrict__ biasp,
                         float* __restrict__ lam) {
  int idx = blockIdx.x * blockDim.x + threadIdx.x;
  if (idx >= M_ * N_ * T_) return;
  int t = idx % T_, mn = idx / T_, m = mn / N_;
  const float* w   = W  + (size_t)mn * L_ * TP + t;
  const float* mup = mu + (size_t)m * L_ * TP + t;
  const float* dhp = dh + (size_t)m * L_ * TP + t;
  float s = biasp[0];
#pragma unroll
  for (int l = 0; l < L_; ++l) {
    const float wv = w[(size_t)l * TP];
    s = fmaf(wv, mup[(size_t)l * TP], s);
    s = fmaf(0.5f * dhp[(size_t)l * TP], wv * wv, s);
  }
  lam[(size_t)mn * TP + t] = __expf(s);
}

// ---- grad1[m,l,t] = sum_n W*(Y-lam);  d[m,l,t] = sum_n W^2*lam ------------
__global__ void k_gradd(const float* __restrict__ W, const float* __restrict__ lam,
                        const float* __restrict__ Y, float* __restrict__ grad,
                        float* __restrict__ dvec) {
  int idx = blockIdx.x * blockDim.x + threadIdx.x;
  if (idx >= MATS * T_) return;
  int t = idx % T_, ml = idx / T_;
  int l = ml % L_, m = ml / L_;
  const float* wb = W   + ((size_t)(m * N_) * L_ + l) * TP + t;  // +n: step L_*TP
  const float* lb = lam + (size_t)(m * N_) * TP + t;             // +n: step TP
  const float* yb = Y   + (size_t)(m * N_) * NTF + NPAD_ + t;    // +n: step NTF
  float g = 0.0f, dd = 0.0f;
  for (int n = 0; n < N_; ++n) {
    const float wv = wb[(size_t)n * L_ * TP];
    const float lv = lb[(size_t)n * TP];
    g  = fmaf(wv, yb[(size_t)n * NTF] - lv, g);
    dd = fmaf(wv * wv, lv, dd);
  }
  grad[(size_t)ml * TP + t] = g;
  dvec[(size_t)ml * TP + t] = dd;
}

// ---- grad -= mu @ invK  via V_WMMA_F32_16X16X4_F32 ------------------------
// M=128 (ml rows), N=K=304.  One wave per 16x16 tile; K-loop in steps of 4.
__global__ __launch_bounds__(32) void k_gemm(const float* __restrict__ mu,
                                             const float* __restrict__ invK,
                                             float* __restrict__ grad) {
  const int lane = threadIdx.x;           // 0..31, wave32
  const int row0 = blockIdx.x * 16;       // 8 blocks  -> 128 rows
  const int col0 = blockIdx.y * 16;       // 19 blocks -> 304 cols
  const int r  = lane & 15;
  const int hi = lane >> 4;
  v8f acc = {};
  const float* arow = mu + (size_t)(row0 + r) * TP + 2 * hi;
  for (int k0 = 0; k0 < TP; k0 += 4) {
    v2f a;
    a.x = arow[k0];
    a.y = arow[k0 + 1];
    v2f b;
    b.x = invK[(size_t)(k0 + 2 * hi + 0) * TP + col0 + r];
    b.y = invK[(size_t)(k0 + 2 * hi + 1) * TP + col0 + r];
    acc = wmma4(a, b, acc);
  }
#pragma unroll
  for (int i = 0; i < 8; ++i) {           // D: VGPR i holds row i + 8*hi
    const size_t o = (size_t)(row0 + i + 8 * hi) * TP + col0 + r;
    grad[o] = grad[o] - acc[i];
  }
}

// ---- A[ml] = invK + diag(d[ml]); identity in padding ----------------------
__global__ void k_buildA(const float* __restrict__ invK, const float* __restrict__ dvec,
                         float* __restrict__ Abuf) {
  const int mat = blockIdx.y;
  const int e = blockIdx.x * blockDim.x + threadIdx.x;
  if (e >= (int)MSZ) return;
  const int i = e / TP, j = e % TP;
  float v;
  if (i < T_ && j < T_) v = invK[e] + (i == j ? dvec[(size_t)mat * TP + i] : 0.0f);
  else                  v = (i == j) ? 1.0f : 0.0f;
  Abuf[(size_t)mat * MSZ + e] = v;
}

// ---- dh = diag(M);  mu -= LR * (M @ grad)   (M = -A^{-1} = hessian^{-1}) --
__global__ void k_finish(const float* __restrict__ Abuf, const float* __restrict__ grad,
                         float* __restrict__ mu, float* __restrict__ dh) {
  const int mat = blockIdx.y;
  const int t = blockIdx.x * blockDim.x + threadIdx.x;
  if (t >= T_) return;
  const float* Mrow = Abuf + (size_t)mat * MSZ + (size_t)t * TP;
  const float* g = grad + (size_t)mat * TP;
  float x = 0.0f;
  for (int s = 0; s < TP; ++s) x = fmaf(Mrow[s], g[s], x);  // pad cols of M are 0
  dh[(size_t)mat * TP + t] = Mrow[t];
  mu[(size_t)mat * TP + t] -= 0.5f * x;   // LR = 0.5
}

// ---- strip padding into output (16, 8, 300) -------------------------------
__global__ void k_out(const float* __restrict__ mu, float* __restrict__ out) {
  int idx = blockIdx.x * blockDim.x + threadIdx.x;
  if (idx >= MATS * T_) return;
  int t = idx % T_, ml = idx / T_;
  out[idx] = mu[(size_t)ml * TP + t];
}

extern "C" void kernel_launch(void* const* d_in, const int* in_sizes, int n_in,
                              void* d_out, int out_size, void* d_ws, size_t ws_size,
                              hipStream_t stream) {
  (void)in_sizes; (void)n_in; (void)out_size; (void)ws_size;
  const float* Y     = (const float*)d_in[0];  // (16,96,350)
  const float* kern  = (const float*)d_in[1];  // (20,8)
  const float* wproj = (const float*)d_in[2];  // (8,8)
  const float* Kin   = (const float*)d_in[3];  // (300,300)
  const float* biasp = (const float*)d_in[4];  // scalar
  float* out = (float*)d_out;

  // workspace layout (floats)
  float* ws   = (float*)d_ws;
  float* invK = ws;                                   // MSZ
  float* Abuf = invK + MSZ;                           // 128 * MSZ
  float* Wbuf = Abuf + (size_t)MATS * MSZ;            // 16*96*8*TP
  float* mu   = Wbuf + (size_t)M_ * N_ * L_ * TP;     // 128*TP
  float* dh   = mu   + (size_t)MATS * TP;
  float* grad = dh   + (size_t)MATS * TP;
  float* dvec = grad + (size_t)MATS * TP;
  float* lam  = dvec + (size_t)MATS * TP;             // 16*96*TP

  const int BLK = 256;
  const int gM  = (int)((MSZ + BLK - 1) / BLK);       // 361

  // precompute: inv_K = inv(K + 1e-3 I) via blocked sweep of matrix slot 0
  k_buildK<<<gM, BLK, 0, stream>>>(Kin, Abuf);
  k_bsweep<<<1, BLK, 0, stream>>>(Abuf);
  k_invk<<<gM, BLK, 0, stream>>>(Abuf, invK);

  // weights (fused conv + projection), state init
  k_weights<<<(M_ * N_ * T_) / BLK, BLK, 0, stream>>>(Y, kern, wproj, Wbuf);
  k_init<<<(MATS * TP) / BLK, BLK, 0, stream>>>(mu, dh);

  for (int it = 0; it < 5; ++it) {
    k_lambda<<<(M_ * N_ * T_) / BLK, BLK, 0, stream>>>(Wbuf, mu, dh, biasp, lam);
    k_gradd<<<(MATS * T_) / BLK, BLK, 0, stream>>>(Wbuf, lam, Y, grad, dvec);
    k_gemm<<<dim3(MATS / 16, TP / 16), 32, 0, stream>>>(mu, invK, grad);
    k_buildA<<<dim3(gM, MATS), BLK, 0, stream>>>(invK, dvec, Abuf);
    k_bsweep<<<MATS, BLK, 0, stream>>>(Abuf);
    k_finish<<<dim3((T_ + BLK - 1) / BLK, MATS), BLK, 0, stream>>>(Abuf, grad, mu, dh);
  }

  k_out<<<(MATS * T_) / BLK, BLK, 0, stream>>>(mu, out);
}